// NNAllpassFilterClone_28226525070332
// MI455X (gfx1250) — compile-verified
//
#include <hip/hip_runtime.h>
#include <stdint.h>

// Reference constants
#define DELAY  2048
#define DELAYV (DELAY / 4)   // float4 slots per delay line = 512
#define BLOCK  256

typedef float v4f __attribute__((ext_vector_type(4)));

// Allpass with delay line: each thread owns 4 consecutive ring-buffer slots of
// one channel and walks its (depth T/DELAY) recurrence chain.
//
//   y_0            = buffer[b, pos]
//   out[b, t]      = y_k - x[b, t]
//   y_{k+1}        = x[b, t] + feedback * y_k        (t = pos + k*DELAY)
//
// Memory-bound: x/out streamed with non-temporal B128 ops; the buffer tile is
// staged HBM -> LDS via the CDNA5 async-to-LDS engine (ASYNCcnt-tracked).
__global__ __launch_bounds__(BLOCK) void NNAllpassFilterClone_kernel(
    const v4f* __restrict__ x4,      // [B, T/4]
    const v4f* __restrict__ buf4,    // [B, DELAY/4]
    const float* __restrict__ fbp,   // scalar feedback
    v4f* __restrict__ out4,          // [B, T/4]
    int TV,                          // T/4
    int n4)                          // B * DELAY/4 (total vec4 slots)
{
    __shared__ v4f tile[BLOCK];

    const int tid = threadIdx.x;
    const int gid = blockIdx.x * BLOCK + tid;
    const bool active = (gid < n4);

    // --- CDNA5 async copy: this block's contiguous 4 KB buffer tile, HBM -> LDS,
    //     bypassing VGPRs. Tracked by ASYNCcnt.
    if (active) {
        unsigned lds_off = (unsigned)(uintptr_t)&tile[tid];              // flat addr low 32b == LDS byte offset
        unsigned long long gaddr = (unsigned long long)(uintptr_t)(buf4 + gid);
        asm volatile("global_load_async_to_lds_b128 %0, %1, off"
                     :: "v"(lds_off), "v"(gaddr)
                     : "memory");
    }
    asm volatile("s_wait_asynccnt 0x0" ::: "memory");
    __syncthreads();

    if (!active) return;

    const float f = *fbp;            // uniform -> scalar load
    v4f y = tile[tid];               // initial ring-buffer state for these 4 slots

    const int b = gid / DELAYV;      // channel
    const int p = gid % DELAYV;      // vec4 slot within the delay line
    const long long base = (long long)b * TV;

    // T/DELAY passes (== 2 for the reference shapes)
    for (int tv = p; tv < TV; tv += DELAYV) {
        v4f xv = __builtin_nontemporal_load(&x4[base + tv]);   // streamed once
        v4f o  = y - xv;                                       // out = -x + buf_out
        __builtin_nontemporal_store(o, &out4[base + tv]);
        y = xv + f * y;                                        // buffer update (FMA)
    }
}

extern "C" void kernel_launch(void* const* d_in, const int* in_sizes, int n_in,
                              void* d_out, int out_size, void* d_ws, size_t ws_size,
                              hipStream_t stream) {
    const v4f*  x4   = (const v4f*)d_in[0];   // x: [B, T] float32
    const v4f*  buf4 = (const v4f*)d_in[1];   // buffer: [B, DELAY] float32
    const float* fb  = (const float*)d_in[2]; // feedback scalar

    v4f* out4 = (v4f*)d_out;

    const int bufN = in_sizes[1];             // B * DELAY
    const int B    = bufN / DELAY;
    const int T    = in_sizes[0] / B;
    const int TV   = T / 4;
    const int n4   = bufN / 4;                // total vec4 delay slots

    const int blocks = (n4 + BLOCK - 1) / BLOCK;
    NNAllpassFilterClone_kernel<<<blocks, BLOCK, 0, stream>>>(x4, buf4, fb, out4, TV, n4);
}